// EmbBGConv_unit_78340203479086
// MI455X (gfx1250) — compile-verified
//
#include <hip/hip_runtime.h>

typedef __attribute__((ext_vector_type(2))) float v2f;
typedef __attribute__((ext_vector_type(4))) float v4f;
typedef __attribute__((ext_vector_type(8))) float v8f;

#define N_OBJ   50000
#define NPAIR   300000
#define DIM     256
#define BASE    10.0f

#define ROW_TILES   (N_OBJ / 16)      // 3125 16-row tiles
#define WAVES_PB    8                 // 8 waves per block (256 threads)
#define KCHUNK      64                // K rows staged per LDS chunk
#define BSTRIDE     80                // LDS row stride (floats): 2*80 = 160 = 32 mod 64
                                      // -> half-wave bank ranges disjoint (64 banks)

// ---------------------------------------------------------------------------
// Stage 0: zero the num accumulator (first N*D floats of d_out), init m=BASE,
// Z=0. Grid-stride so one launch covers everything.
// ---------------------------------------------------------------------------
__global__ void init_kernel(float* __restrict__ num, float* __restrict__ m,
                            float* __restrict__ Z) {
    const long long total = (long long)N_OBJ * DIM;
    for (long long i = (long long)blockIdx.x * blockDim.x + threadIdx.x;
         i < total; i += (long long)gridDim.x * blockDim.x) {
        num[i] = 0.0f;
        if (i < N_OBJ) { m[i] = BASE; Z[i] = 0.0f; }
    }
}

// ---------------------------------------------------------------------------
// Stage 1: emb = feats @ W + b via V_WMMA_F32_16X16X4_F32.
// Block = 8 waves; wave w computes rows [ (bx*8+w)*16 , +16 ), cols [by*64,+64).
// W slab (KCHUNK x 64 cols) is cooperatively staged in LDS per K-chunk and
// B fragments are served by ds_load; A fragments come straight from global
// (pairs merge into global_load_b64).
// Fragment layouts (wave32):
//   A 16x4 f32 (2 VGPR): lane l -> row (l&15), K = 2*(l>>4)+{0,1}
//   B 4x16 f32 (2 VGPR): lane l -> col (l&15), K = 2*(l>>4)+{0,1}
//   C/D 16x16 f32 (8 VGPR): vgpr i -> row i + 8*(l>>4), col (l&15)
// ---------------------------------------------------------------------------
__global__ __launch_bounds__(32 * WAVES_PB)
void gemm_wmma_kernel(const float* __restrict__ feats,
                      const float* __restrict__ W,
                      const float* __restrict__ bias,
                      float* __restrict__ emb) {
    __shared__ float Bs[KCHUNK * BSTRIDE];   // 20 KB of the 320 KB WGP LDS

    const int tid  = threadIdx.x;
    const int wid  = tid >> 5;         // wave id 0..7
    const int lane = tid & 31;
    const int hh   = lane >> 4;        // half-wave select (K pair)
    const int lm   = lane & 15;

    const int tile = blockIdx.x * WAVES_PB + wid;   // 16-row tile index
    const bool active = tile < ROW_TILES;
    const int row0 = tile * 16;
    const int col0 = blockIdx.y * 64;

    v8f acc0 = {}, acc1 = {}, acc2 = {}, acc3 = {};

    const float* arow = active ? feats + (size_t)(row0 + lm) * DIM + 2 * hh
                               : feats + 2 * hh;   // harmless fallback

    // cooperative-load mapping: 64 rows x 64 cols, 4 threads per row,
    // each thread moves 4 float4s (16 floats)
    const int lrow = tid >> 2;         // 0..63
    const int lq   = tid & 3;          // 0..3  -> cols lq*16 .. lq*16+15

    for (int kc = 0; kc < DIM; kc += KCHUNK) {
        // ---- stage W[kc..kc+63][col0..col0+63] into LDS ----
        {
            const float* wsrc = W + (size_t)(kc + lrow) * DIM + col0 + lq * 16;
            float* ldst = Bs + lrow * BSTRIDE + lq * 16;
            #pragma unroll
            for (int j = 0; j < 4; ++j) {
                v4f v = *(const v4f*)(wsrc + 4 * j);
                *(v4f*)(ldst + 4 * j) = v;
            }
        }
        __syncthreads();

        if (active) {
            #pragma unroll 4
            for (int k = 0; k < KCHUNK; k += 4) {
                v2f a;
                a.x = arow[kc + k];
                a.y = arow[kc + k + 1];

                const float* b0p = Bs + (k + 2 * hh) * BSTRIDE + lm;
                const float* b1p = b0p + BSTRIDE;   // next K row
                v2f b0; b0.x = b0p[0];  b0.y = b1p[0];
                v2f b1; b1.x = b0p[16]; b1.y = b1p[16];
                v2f b2; b2.x = b0p[32]; b2.y = b1p[32];
                v2f b3; b3.x = b0p[48]; b3.y = b1p[48];

                acc0 = __builtin_amdgcn_wmma_f32_16x16x4_f32(false, a, false, b0,
                                                             (short)0, acc0, false, false);
                acc1 = __builtin_amdgcn_wmma_f32_16x16x4_f32(false, a, false, b1,
                                                             (short)0, acc1, false, false);
                acc2 = __builtin_amdgcn_wmma_f32_16x16x4_f32(false, a, false, b2,
                                                             (short)0, acc2, false, false);
                acc3 = __builtin_amdgcn_wmma_f32_16x16x4_f32(false, a, false, b3,
                                                             (short)0, acc3, false, false);
            }
        }
        __syncthreads();
    }

    if (!active) return;

    // Write back D tiles with bias (bias depends only on the column).
    const float bb0 = bias[col0 + lm];
    const float bb1 = bias[col0 + 16 + lm];
    const float bb2 = bias[col0 + 32 + lm];
    const float bb3 = bias[col0 + 48 + lm];
    #pragma unroll
    for (int i = 0; i < 8; ++i) {
        const int row = row0 + i + 8 * hh;
        float* orow = emb + (size_t)row * DIM + col0 + lm;
        orow[0]  = acc0[i] + bb0;
        orow[16] = acc1[i] + bb1;
        orow[32] = acc2[i] + bb2;
        orow[48] = acc3[i] + bb3;
    }
}

// ---------------------------------------------------------------------------
// Stage 2: segment max of confidence over sub/obj. m pre-initialized to BASE
// (positive), so signed-int atomicMax is order-correct: positive floats
// compare correctly as ints and negative candidates can never exceed +10.
// ---------------------------------------------------------------------------
__global__ void segmax_kernel(const long long* __restrict__ pairs,
                              const float* __restrict__ conf,
                              float* __restrict__ m) {
    for (int p = blockIdx.x * blockDim.x + threadIdx.x; p < NPAIR;
         p += gridDim.x * blockDim.x) {
        const int s = (int)pairs[2 * p];
        const int o = (int)pairs[2 * p + 1];
        const int cv = __float_as_int(conf[p]);
        atomicMax((int*)(m + s), cv);
        atomicMax((int*)(m + o), cv);
    }
}

// ---------------------------------------------------------------------------
// Stage 3: scatter-accumulate. One 256-thread block per pair; thread t owns
// dim t. num lives in the first N*D floats of d_out. emb + num are L2-resident
// (51 MB each << 192 MB L2), so the 2 gathers + 2 fp32 atomics per element
// run at L2 rates.
// ---------------------------------------------------------------------------
__global__ __launch_bounds__(DIM)
void scatter_kernel(const long long* __restrict__ pairs,
                    const float* __restrict__ conf,
                    const float* __restrict__ emb,
                    const float* __restrict__ m,
                    float* __restrict__ num,
                    float* __restrict__ Z) {
    const int p = blockIdx.x;
    const long long s = pairs[2 * p];
    const long long o = pairs[2 * p + 1];
    const float c  = conf[p];
    const float ws = __expf(c - m[s]);
    const float wo = __expf(c - m[o]);

    const int d = threadIdx.x;
    const float es = emb[(size_t)s * DIM + d];
    const float eo = emb[(size_t)o * DIM + d];
    const float prod = es * eo;

    atomicAdd(num + (size_t)s * DIM + d, ws * prod);
    atomicAdd(num + (size_t)o * DIM + d, wo * prod);
    if (d == 0) {
        atomicAdd(Z + s, ws);
        atomicAdd(Z + o, wo);
    }
}

// ---------------------------------------------------------------------------
// Stage 4: finalize: new_feats = (num + w_self*feats) / (Z + w_self)
// ---------------------------------------------------------------------------
__global__ __launch_bounds__(DIM)
void finalize_kernel(const float* __restrict__ feats,
                     const float* __restrict__ m,
                     const float* __restrict__ Z,
                     float* __restrict__ out) {
    const int i = blockIdx.x;
    const int d = threadIdx.x;
    const float wself = __expf(BASE - m[i]);
    const float z = Z[i] + wself;
    const size_t idx = (size_t)i * DIM + d;
    out[idx] = (out[idx] + wself * feats[idx]) / z;
}

// ---------------------------------------------------------------------------
// Stage 5: tuple tail — pairs (cast to float, lossless: indices < 2^24) and
// confidence, concatenated after new_feats.
// ---------------------------------------------------------------------------
__global__ void tail_kernel(const long long* __restrict__ pairs,
                            const float* __restrict__ conf,
                            float* __restrict__ out) {
    const size_t base = (size_t)N_OBJ * DIM;
    for (int p = blockIdx.x * blockDim.x + threadIdx.x; p < NPAIR;
         p += gridDim.x * blockDim.x) {
        out[base + 2 * p]     = (float)pairs[2 * p];
        out[base + 2 * p + 1] = (float)pairs[2 * p + 1];
        out[base + (size_t)2 * NPAIR + p] = conf[p];
    }
}

// ---------------------------------------------------------------------------
extern "C" void kernel_launch(void* const* d_in, const int* in_sizes, int n_in,
                              void* d_out, int out_size, void* d_ws, size_t ws_size,
                              hipStream_t stream) {
    const float*      feats = (const float*)d_in[0];      // [N, D] f32
    const long long*  pairs = (const long long*)d_in[1];  // [P, 2] i64
    const float*      conf  = (const float*)d_in[2];      // [P]    f32
    const float*      W     = (const float*)d_in[3];      // [D, D] f32
    const float*      bias  = (const float*)d_in[4];      // [D]    f32
    float*            out   = (float*)d_out;

    // workspace: emb [N*D] | m [N] | Z [N]
    float* emb = (float*)d_ws;
    float* m   = emb + (size_t)N_OBJ * DIM;
    float* Z   = m + N_OBJ;

    // 0: zero num (d_out accumulator region), init m=BASE, Z=0
    init_kernel<<<4096, 256, 0, stream>>>(out, m, Z);

    // 1: emb = feats @ W + b  (fp32 WMMA, 8 waves/block, LDS-staged W slab)
    const int gx = (ROW_TILES + WAVES_PB - 1) / WAVES_PB;   // 391
    gemm_wmma_kernel<<<dim3(gx, DIM / 64), 32 * WAVES_PB, 0, stream>>>(feats, W, bias, emb);

    // 2: per-node confidence max (with BASELINE folded into init)
    segmax_kernel<<<2048, 256, 0, stream>>>(pairs, conf, m);

    // 3: softmax-weighted scatter of emb[sub]*emb[obj]
    scatter_kernel<<<NPAIR, DIM, 0, stream>>>(pairs, conf, emb, m, out, Z);

    // 4: normalize + self term
    finalize_kernel<<<N_OBJ, DIM, 0, stream>>>(feats, m, Z, out);

    // 5: pass-through tuple outputs
    tail_kernel<<<2048, 256, 0, stream>>>(pairs, conf, out);
}